// APP_81192061764217
// MI455X (gfx1250) — compile-verified
//
#include <hip/hip_runtime.h>

// Problem constants (from reference setup_inputs)
#define NN 50000
#define EE 600000
#define FD 128
#define H1 256
#define H2 128
#define CC 40
#define CCP 48            // Wc columns padded to 3 n-tiles
#define ALPHA 0.1f

typedef __attribute__((ext_vector_type(2))) float v2f;
typedef __attribute__((ext_vector_type(8))) float v8f;

// D = A(16x4) * B(4x16) + C(16x16), all f32, wave32.
__device__ __forceinline__ v8f wmma4(v2f a, v2f b, v8f c) {
    return __builtin_amdgcn_wmma_f32_16x16x4_f32(
        /*neg_a=*/false, a, /*neg_b=*/false, b,
        /*c_mod=*/(short)0, c, /*reuse_a=*/false, /*reuse_b=*/false);
}

__device__ __forceinline__ float wave_sum32(float s) {
    s += __shfl_xor(s, 16, 32);
    s += __shfl_xor(s, 8, 32);
    s += __shfl_xor(s, 4, 32);
    s += __shfl_xor(s, 2, 32);
    s += __shfl_xor(s, 1, 32);
    return s;
}

// ---------------------------------------------------------------------------
// Weight transposes (one-time, ~72K elements): W1t[n][k], W2t[n][k], Wct[n][k]
// (B fragment for lane = two consecutive K at fixed N -> single b64 load)
// ---------------------------------------------------------------------------
__global__ __launch_bounds__(256)
void prep_kernel(const float* __restrict__ W1, const float* __restrict__ W2,
                 const float* __restrict__ Wc,
                 float* __restrict__ W1t, float* __restrict__ W2t,
                 float* __restrict__ Wct)
{
    int i = blockIdx.x * 256 + threadIdx.x;
    if (i < H1 * FD) {                               // 32768: W1t[n*FD+k]
        int n = i >> 7, k = i & 127;
        W1t[i] = W1[k * H1 + n];
    } else if (i < H1 * FD + H2 * H1) {              // 32768: W2t[n*H1+k]
        int j = i - H1 * FD;
        int n = j >> 8, k = j & 255;
        W2t[j] = W2[k * H2 + n];
    } else if (i < H1 * FD + H2 * H1 + CCP * H2) {   // 6144: Wct[n*H2+k], zero-pad n>=40
        int j = i - H1 * FD - H2 * H1;
        int n = j >> 7, k = j & 127;
        Wct[j] = (n < CC) ? Wc[k * CC + n] : 0.f;
    }
}

// ---------------------------------------------------------------------------
// Edge kernel: per 64-edge tile (LDS ~100.6 KB, probing the 320KB/WG limit)
//   nb_norm = l2norm(neighbor_x)        (LDS, stride 132 -> conflict-free b64)
//   n_h     = nb_norm @ W1 + b1         (WMMA f32; B cached in regs per n-tile)
//   S1 += n_h ; S2 += relu(n_h)         (L2-resident f32 atomics)
//   n_h2    = n_h @ W2 + b2             (WMMA f32 from LDS n_h)
//   S3 += n_h2 ; S4 += relu(n_h2)
// ---------------------------------------------------------------------------
__global__ __launch_bounds__(128)
void edge_kernel(const float* __restrict__ nb,
                 const float* __restrict__ W1t, const float* __restrict__ b1,
                 const float* __restrict__ W2t, const float* __restrict__ b2,
                 const int*   __restrict__ segids,
                 float* __restrict__ S1, float* __restrict__ S2,
                 float* __restrict__ S3, float* __restrict__ S4)
{
    __shared__ float Xs[64 * 132];    // 64 x 128 (pad->132; 132%64==4 conflict-free)
    __shared__ float NHs[64 * 260];   // 64 x 256 (pad->260)
    __shared__ float invn[64];
    __shared__ int   segs[64];

    const int tid  = threadIdx.x;
    const int lane = tid & 31;
    const int w    = tid >> 5;
    const int e0   = blockIdx.x * 64;

    // ---- load 64 x 128 neighbor rows (float4, coalesced) ----
    for (int i = tid; i < 64 * 32; i += 128) {
        int r = i >> 5, c4 = (i & 31) << 2;
        const float4 v = *(const float4*)(nb + (size_t)(e0 + r) * FD + c4);
        float* d = &Xs[r * 132 + c4];
        d[0] = v.x; d[1] = v.y; d[2] = v.z; d[3] = v.w;
    }
    if (tid < 64) segs[tid] = segids[e0 + tid];
    __syncthreads();

    // ---- per-row L2 norm (wave w owns rows 16w..16w+15) ----
    for (int rr = 0; rr < 16; ++rr) {
        int r = w * 16 + rr;
        float s = 0.f;
        for (int c = lane; c < FD; c += 32) { float t = Xs[r * 132 + c]; s += t * t; }
        s = wave_sum32(s);
        if (lane == 0) invn[r] = 1.0f / fmaxf(sqrtf(s), 1e-12f);
    }
    __syncthreads();
    for (int i = tid; i < 64 * FD; i += 128) {
        int r = i >> 7, c = i & 127;
        Xs[r * 132 + c] *= invn[r];
    }
    __syncthreads();

    const int nlo = lane & 15;          // N index within tile / M index for A
    const int kb  = (lane >> 4) << 1;   // K sub-offset for A/B fragments
    const int rb  = (lane >> 4) << 3;   // M sub-offset for C/D rows

    // ---- GEMM1: [64x128] @ W1[128x256]; wave w owns n-tiles 4w..4w+3 ----
    for (int nn = 0; nn < 4; ++nn) {
        int n0 = (w * 4 + nn) * 16;
        v2f Bc[32];                     // full-K B cache, reused by 4 m-tiles
        #pragma unroll
        for (int k4 = 0; k4 < 32; ++k4)
            Bc[k4] = *(const v2f*)&W1t[(n0 + nlo) * FD + k4 * 4 + kb];
        float bias = b1[n0 + nlo];
        for (int mt = 0; mt < 4; ++mt) {
            v8f acc = {0.f, 0.f, 0.f, 0.f, 0.f, 0.f, 0.f, 0.f};
            #pragma unroll
            for (int k4 = 0; k4 < 32; ++k4) {
                v2f a;
                const float* ap = &Xs[(mt * 16 + nlo) * 132 + k4 * 4 + kb];
                a[0] = ap[0]; a[1] = ap[1];
                acc = wmma4(a, Bc[k4], acc);
            }
            #pragma unroll
            for (int r = 0; r < 8; ++r) {
                int row = mt * 16 + rb + r;            // D: VGPR r -> M=r / r+8
                float v = acc[r] + bias;
                NHs[row * 260 + n0 + nlo] = v;         // pre-ReLU n_h feeds GEMM2
                int sg = segs[row];
                atomicAdd(&S1[(size_t)sg * H1 + n0 + nlo], v);
                atomicAdd(&S2[(size_t)sg * H1 + n0 + nlo], fmaxf(v, 0.f));
            }
        }
    }
    __syncthreads();

    // ---- GEMM2: n_h[64x256] @ W2[256x128]; wave w owns n-tiles 2w..2w+1 ----
    for (int nn = 0; nn < 2; ++nn) {
        int n0 = (w * 2 + nn) * 16;
        v8f accs[4];
        #pragma unroll
        for (int mt = 0; mt < 4; ++mt)
            accs[mt] = (v8f){0.f, 0.f, 0.f, 0.f, 0.f, 0.f, 0.f, 0.f};
        for (int kc = 0; kc < 2; ++kc) {               // K split: 2 x 128
            v2f Bc[32];
            #pragma unroll
            for (int k4 = 0; k4 < 32; ++k4)
                Bc[k4] = *(const v2f*)&W2t[(n0 + nlo) * H1 + kc * 128 + k4 * 4 + kb];
            for (int mt = 0; mt < 4; ++mt) {
                #pragma unroll
                for (int k4 = 0; k4 < 32; ++k4) {
                    v2f a;
                    const float* ap = &NHs[(mt * 16 + nlo) * 260 + kc * 128 + k4 * 4 + kb];
                    a[0] = ap[0]; a[1] = ap[1];
                    accs[mt] = wmma4(a, Bc[k4], accs[mt]);
                }
            }
        }
        float bias = b2[n0 + nlo];
        for (int mt = 0; mt < 4; ++mt) {
            #pragma unroll
            for (int r = 0; r < 8; ++r) {
                int row = mt * 16 + rb + r;
                float v = accs[mt][r] + bias;
                int sg = segs[row];
                atomicAdd(&S3[(size_t)sg * H2 + n0 + nlo], v);
                atomicAdd(&S4[(size_t)sg * H2 + n0 + nlo], fmaxf(v, 0.f));
            }
        }
    }
}

// ---------------------------------------------------------------------------
// Node kernel 1: h = l2norm(x) @ W1 + b1   (32-row tiles, guarded)
// ---------------------------------------------------------------------------
__global__ __launch_bounds__(128)
void node1_kernel(const float* __restrict__ x,
                  const float* __restrict__ W1t, const float* __restrict__ b1,
                  float* __restrict__ hout)
{
    __shared__ float Xs[32 * 132];
    __shared__ float invn[32];

    const int tid  = threadIdx.x;
    const int lane = tid & 31;
    const int w    = tid >> 5;
    const int r0   = blockIdx.x * 32;

    for (int i = tid; i < 32 * 32; i += 128) {
        int r = i >> 5, c4 = (i & 31) << 2;
        int g = r0 + r;
        float4 v = make_float4(0.f, 0.f, 0.f, 0.f);
        if (g < NN) v = *(const float4*)(x + (size_t)g * FD + c4);
        float* d = &Xs[r * 132 + c4];
        d[0] = v.x; d[1] = v.y; d[2] = v.z; d[3] = v.w;
    }
    __syncthreads();

    for (int rr = 0; rr < 8; ++rr) {
        int r = w * 8 + rr;
        float s = 0.f;
        for (int c = lane; c < FD; c += 32) { float t = Xs[r * 132 + c]; s += t * t; }
        s = wave_sum32(s);
        if (lane == 0) invn[r] = 1.0f / fmaxf(sqrtf(s), 1e-12f);
    }
    __syncthreads();
    for (int i = tid; i < 32 * FD; i += 128) {
        int r = i >> 7, c = i & 127;
        Xs[r * 132 + c] *= invn[r];
    }
    __syncthreads();

    const int nlo = lane & 15;
    const int kb  = (lane >> 4) << 1;
    const int rb  = (lane >> 4) << 3;

    // wave w owns n-tiles 4w..4w+3, 2 m-tiles each
    for (int nn = 0; nn < 4; ++nn) {
        int n0 = (w * 4 + nn) * 16;
        v2f Bc[32];
        #pragma unroll
        for (int k4 = 0; k4 < 32; ++k4)
            Bc[k4] = *(const v2f*)&W1t[(n0 + nlo) * FD + k4 * 4 + kb];
        float bias = b1[n0 + nlo];
        for (int mt = 0; mt < 2; ++mt) {
            v8f acc = {0.f, 0.f, 0.f, 0.f, 0.f, 0.f, 0.f, 0.f};
            #pragma unroll
            for (int k4 = 0; k4 < 32; ++k4) {
                v2f a;
                const float* ap = &Xs[(mt * 16 + nlo) * 132 + k4 * 4 + kb];
                a[0] = ap[0]; a[1] = ap[1];
                acc = wmma4(a, Bc[k4], acc);
            }
            #pragma unroll
            for (int r = 0; r < 8; ++r) {
                int g = r0 + mt * 16 + rb + r;
                if (g < NN) hout[(size_t)g * H1 + n0 + nlo] = acc[r] + bias;
            }
        }
    }
}

// ---------------------------------------------------------------------------
// Node kernel 2: x2 = app2(app1(h,S1),S2,h) on the fly; h2 = x2 @ W2 + b2
// ---------------------------------------------------------------------------
__global__ __launch_bounds__(128)
void node2_kernel(const float* __restrict__ h,
                  const float* __restrict__ S1, const float* __restrict__ S2,
                  const float* __restrict__ W2t, const float* __restrict__ b2,
                  float* __restrict__ h2out)
{
    __shared__ float X2s[32 * 260];

    const int tid  = threadIdx.x;
    const int lane = tid & 31;
    const int w    = tid >> 5;
    const int r0   = blockIdx.x * 32;

    for (int i = tid; i < 32 * H1; i += 128) {
        int r = i >> 8, c = i & 255;
        int g = r0 + r;
        float v = 0.f;
        if (g < NN) {
            size_t o = (size_t)g * H1 + c;
            float hv = h[o];
            float x1 = fmaxf((1.f - ALPHA) * (hv + S1[o]) + ALPHA * hv, 0.f);
            v = (1.f - ALPHA) * (x1 + S2[o]) + ALPHA * hv;
        }
        X2s[r * 260 + c] = v;
    }
    __syncthreads();

    const int nlo = lane & 15;
    const int kb  = (lane >> 4) << 1;
    const int rb  = (lane >> 4) << 3;

    // wave w owns n-tiles 2w..2w+1, 2 m-tiles, K=256 split in 2
    for (int nn = 0; nn < 2; ++nn) {
        int n0 = (w * 2 + nn) * 16;
        v8f accs[2];
        accs[0] = (v8f){0.f, 0.f, 0.f, 0.f, 0.f, 0.f, 0.f, 0.f};
        accs[1] = (v8f){0.f, 0.f, 0.f, 0.f, 0.f, 0.f, 0.f, 0.f};
        for (int kc = 0; kc < 2; ++kc) {
            v2f Bc[32];
            #pragma unroll
            for (int k4 = 0; k4 < 32; ++k4)
                Bc[k4] = *(const v2f*)&W2t[(n0 + nlo) * H1 + kc * 128 + k4 * 4 + kb];
            for (int mt = 0; mt < 2; ++mt) {
                #pragma unroll
                for (int k4 = 0; k4 < 32; ++k4) {
                    v2f a;
                    const float* ap = &X2s[(mt * 16 + nlo) * 260 + kc * 128 + k4 * 4 + kb];
                    a[0] = ap[0]; a[1] = ap[1];
                    accs[mt] = wmma4(a, Bc[k4], accs[mt]);
                }
            }
        }
        float bias = b2[n0 + nlo];
        for (int mt = 0; mt < 2; ++mt) {
            #pragma unroll
            for (int r = 0; r < 8; ++r) {
                int g = r0 + mt * 16 + rb + r;
                if (g < NN) h2out[(size_t)g * H2 + n0 + nlo] = accs[mt][r] + bias;
            }
        }
    }
}

// ---------------------------------------------------------------------------
// Node kernel 3: x4 from (h2,S3,S4); out = x4 @ Wct + bc  (48 padded cols)
// ---------------------------------------------------------------------------
__global__ __launch_bounds__(128)
void node3_kernel(const float* __restrict__ h2,
                  const float* __restrict__ S3, const float* __restrict__ S4,
                  const float* __restrict__ Wct, const float* __restrict__ bc,
                  float* __restrict__ out)
{
    __shared__ float X4s[32 * 132];

    const int tid  = threadIdx.x;
    const int lane = tid & 31;
    const int w    = tid >> 5;
    const int r0   = blockIdx.x * 32;

    for (int i = tid; i < 32 * H2; i += 128) {
        int r = i >> 7, c = i & 127;
        int g = r0 + r;
        float v = 0.f;
        if (g < NN) {
            size_t o = (size_t)g * H2 + c;
            float hv = h2[o];
            float x3 = fmaxf((1.f - ALPHA) * (hv + S3[o]) + ALPHA * hv, 0.f);
            v = (1.f - ALPHA) * (x3 + S4[o]) + ALPHA * hv;
        }
        X4s[r * 132 + c] = v;
    }
    __syncthreads();

    const int nlo = lane & 15;
    const int kb  = (lane >> 4) << 1;
    const int rb  = (lane >> 4) << 3;

    // 2 m-tiles x 3 n-tiles = 6 pairs over 4 waves (padded cols are zero in Wct)
    for (int p = w; p < 6; p += 4) {
        int nt = p >> 1, mt = p & 1;
        int n0 = nt * 16;
        int col = n0 + nlo;
        v8f acc = {0.f, 0.f, 0.f, 0.f, 0.f, 0.f, 0.f, 0.f};
        #pragma unroll
        for (int k4 = 0; k4 < 32; ++k4) {
            v2f a;
            const float* ap = &X4s[(mt * 16 + nlo) * 132 + k4 * 4 + kb];
            a[0] = ap[0]; a[1] = ap[1];
            v2f b = *(const v2f*)&Wct[col * H2 + k4 * 4 + kb];
            acc = wmma4(a, b, acc);
        }
        bool cok = (col < CC);
        float bias = cok ? bc[col] : 0.f;
        #pragma unroll
        for (int r = 0; r < 8; ++r) {
            int g = r0 + mt * 16 + rb + r;
            if (g < NN && cok) out[(size_t)g * CC + col] = acc[r] + bias;
        }
    }
}

// ---------------------------------------------------------------------------
extern "C" void kernel_launch(void* const* d_in, const int* in_sizes, int n_in,
                              void* d_out, int out_size, void* d_ws, size_t ws_size,
                              hipStream_t stream)
{
    const float* x   = (const float*)d_in[0];
    const float* nb  = (const float*)d_in[1];
    const float* W1  = (const float*)d_in[2];
    const float* b1  = (const float*)d_in[3];
    const float* W2  = (const float*)d_in[4];
    const float* b2  = (const float*)d_in[5];
    const float* Wc  = (const float*)d_in[6];
    const float* bc  = (const float*)d_in[7];
    const int*   seg = (const int*)d_in[8];
    float* out = (float*)d_out;

    float* wsf = (float*)d_ws;
    float* S1  = wsf;                       // N x H1
    float* S2  = S1  + (size_t)NN * H1;     // N x H1
    float* S3  = S2  + (size_t)NN * H1;     // N x H2
    float* S4  = S3  + (size_t)NN * H2;     // N x H2
    float* hb  = S4  + (size_t)NN * H2;     // N x H1
    float* h2  = hb  + (size_t)NN * H1;     // N x H2
    float* W1t = h2  + (size_t)NN * H2;     // H1 x FD
    float* W2t = W1t + (size_t)H1 * FD;     // H2 x H1
    float* Wct = W2t + (size_t)H2 * H1;     // CCP x H2

    // zero segment accumulators (153.6 MB -> L2 resident)
    hipMemsetAsync(S1, 0, (size_t)NN * (H1 + H1 + H2 + H2) * sizeof(float), stream);

    const int prepBlocks = (H1 * FD + H2 * H1 + CCP * H2 + 255) / 256;  // 280
    const int nodeBlocks = (NN + 31) / 32;   // 1563
    const int edgeBlocks = EE / 64;          // 9375

    prep_kernel <<<prepBlocks, 256, 0, stream>>>(W1, W2, Wc, W1t, W2t, Wct);
    node1_kernel<<<nodeBlocks, 128, 0, stream>>>(x, W1t, b1, hb);
    edge_kernel <<<edgeBlocks, 128, 0, stream>>>(nb, W1t, b1, W2t, b2, seg, S1, S2, S3, S4);
    node2_kernel<<<nodeBlocks, 128, 0, stream>>>(hb, S1, S2, W2t, b2, h2);
    node3_kernel<<<nodeBlocks, 128, 0, stream>>>(h2, S3, S4, Wct, bc, out);
}